// ActorNetwork_76476187672806
// MI455X (gfx1250) — compile-verified
//
#include <hip/hip_runtime.h>

typedef float v2f __attribute__((ext_vector_type(2)));
typedef float v8f __attribute__((ext_vector_type(8)));

#define DEV __device__ __forceinline__

// Order LDS writes vs subsequent cross-lane LDS reads within one wave.
DEV void lds_sync_wave() { asm volatile("s_wait_dscnt 0" ::: "memory"); }
// Wait for async global<->LDS transfers issued by this wave.
DEV void async_sync_wave() { asm volatile("s_wait_asynccnt 0" ::: "memory"); }

// LDS byte offset of a generic pointer into __shared__ memory.
typedef __attribute__((address_space(3))) char lds_char;
DEV unsigned ldsOffsetOf(const void* p) {
  return (unsigned)(size_t)(lds_char*)p;
}

DEV void async_load_b128(unsigned ldsOff, const void* gaddr) {
  asm volatile("global_load_async_to_lds_b128 %0, %1, off"
               :: "v"(ldsOff), "v"(gaddr) : "memory");
}
DEV void async_load_b32(unsigned ldsOff, const void* gaddr) {
  asm volatile("global_load_async_to_lds_b32 %0, %1, off"
               :: "v"(ldsOff), "v"(gaddr) : "memory");
}
DEV void async_store_b128(void* gaddr, unsigned ldsOff) {
  asm volatile("global_store_async_from_lds_b128 %0, %1, off"
               :: "v"(gaddr), "v"(ldsOff) : "memory");
}

DEV v8f wmma_f32_4(v2f a, v2f b, v8f c) {
  // D = A(16x4, f32) * B(4x16, f32) + C(16x16, f32)
  return __builtin_amdgcn_wmma_f32_16x16x4_f32(false, a, false, b, (short)0, c,
                                               false, false);
}

struct MLP3 { const float *W1,*b1,*W2,*b2,*W3,*b3; };

// One 16-row tile  x  [Kreal x Nout] weight, via V_WMMA_F32_16X16X4_F32.
// ldsIn: 16 rows, strideIn floats, zero-padded cols [Kreal, Kpad).
// A layout: lane holds row M=lane&15; lanes<16 carry K {k,k+1}, lanes>=16 {k+2,k+3}.
// B layout (mirror): lane holds col N=lane&15, same K split.
// C layout: acc[v] -> row (v + 8*(lane>=16)), col lane&15.
DEV void tile_layer(const float* ldsIn, int strideIn, int Kpad, int Kreal,
                    const float* Wg, const float* bias, int Nout,
                    float* ldsOut, int strideOut, bool relu, int lane)
{
  const int M0    = lane & 15;
  const int koff  = (lane < 16) ? 0 : 2;
  const int mbase = (lane < 16) ? 0 : 8;
  const int ncl   = lane & 15;
  const int ntiles = (Nout + 15) >> 4;
  for (int nt = 0; nt < ntiles; ++nt) {
    const int nc = nt * 16 + ncl;
    v8f acc = {0.f,0.f,0.f,0.f,0.f,0.f,0.f,0.f};
    for (int k = 0; k < Kpad; k += 4) {
      const int kk = k + koff;
      v2f a, b;
      a.x = ldsIn[M0 * strideIn + kk];
      a.y = ldsIn[M0 * strideIn + kk + 1];
      b.x = (nc < Nout && kk     < Kreal) ? Wg[kk * Nout + nc]       : 0.f;
      b.y = (nc < Nout && kk + 1 < Kreal) ? Wg[(kk + 1) * Nout + nc] : 0.f;
      acc = wmma_f32_4(a, b, acc);
    }
    if (nc < Nout) {
      const float bv = bias[nc];
#pragma unroll
      for (int v = 0; v < 8; ++v) {
        float val = acc[v] + bv;
        if (relu) val = fmaxf(val, 0.f);
        ldsOut[(mbase + v) * strideOut + nc] = val;
      }
    }
  }
  lds_sync_wave();
}

// ---------------- degree / normalization ----------------
__global__ __launch_bounds__(256) void k_deg_init(float* deg, int N) {
  int i = blockIdx.x * 256 + threadIdx.x;
  if (i < N) deg[i] = 1.0f;                    // self loop contributes 1
}
__global__ __launch_bounds__(256) void k_deg_edges(const int* colI, float* deg, int E) {
  int e = blockIdx.x * 256 + threadIdx.x;
  if (e < E) atomicAdd(&deg[colI[e]], 1.0f);
}
__global__ __launch_bounds__(256) void k_dinv(const float* deg, float* dinv, int N) {
  int i = blockIdx.x * 256 + threadIdx.x;
  if (i < N) { float d = deg[i]; dinv[i] = (d > 0.f) ? 1.0f / sqrtf(d) : 0.f; }
}

// ---------------- mlp1: x[N,5] -> h[N,16] ----------------
__global__ __launch_bounds__(256) void k_mlp1(const float* __restrict__ x, MLP3 p,
                                              float* __restrict__ h, int N) {
  __shared__ float lds[8][2][16 * 32];
  const int wave = threadIdx.x >> 5, lane = threadIdx.x & 31;
  const int rowBase = blockIdx.x * 128 + wave * 16;
  if (rowBase >= N) return;
  float* bufA = lds[wave][0]; float* bufB = lds[wave][1];
  const bool full = (rowBase + 16 <= N);
  if (full) {
    // async gather 16 rows x 5 floats into stride-32 LDS tile; zero K-pad cols
    const unsigned lb = ldsOffsetOf(bufA);
#pragma unroll
    for (int j = 0; j < 3; ++j) {
      int idx = j * 32 + lane;
      if (idx < 80) {
        int row = idx / 5, col = idx % 5;
        async_load_b32(lb + (unsigned)(row * 128 + col * 4),
                       x + (size_t)(rowBase + row) * 5 + col);
      }
    }
    for (int idx = lane; idx < 48; idx += 32) {     // cols 5..7 = 0
      int row = idx / 3, c = 5 + idx % 3;
      bufA[row * 32 + c] = 0.f;
    }
  } else {
    for (int idx = lane; idx < 16 * 8; idx += 32) {
      int m = idx >> 3, c = idx & 7, r = rowBase + m;
      bufA[m * 32 + c] = (r < N && c < 5) ? x[r * 5 + c] : 0.f;
    }
  }
  lds_sync_wave();
  async_sync_wave();
  tile_layer(bufA, 32, 8, 5,  p.W1, p.b1, 32, bufB, 32, true,  lane);
  tile_layer(bufB, 32, 32, 32, p.W2, p.b2, 16, bufA, 32, true,  lane);
  tile_layer(bufA, 32, 16, 16, p.W3, p.b3, 16, bufB, 32, false, lane);
  if (full) {
    // async store 16x16 tile (1024B contiguous): 2 x b128, lane -> 16B chunk
    const unsigned lb = ldsOffsetOf(bufB);
#pragma unroll
    for (int q = 0; q < 2; ++q) {
      int t = q * 32 + lane;                        // 16-byte chunk index
      int row = t >> 2, colb = (t & 3) * 16;
      async_store_b128((char*)(h + (size_t)rowBase * 16) + t * 16,
                       lb + (unsigned)(row * 128 + colb));
    }
    async_sync_wave();
  } else {
    for (int idx = lane; idx < 16 * 16; idx += 32) {
      int m = idx >> 4, c = idx & 15, r = rowBase + m;
      if (r < N) h[r * 16 + c] = bufB[m * 32 + c];
    }
  }
}

// ---------------- edge aggregation ----------------
__global__ __launch_bounds__(256) void k_aggr_init(const float* h, const float* dinv,
                                                   float* aggr, int N) {
  int idx = blockIdx.x * 256 + threadIdx.x;
  if (idx < N * 16) {
    int i = idx >> 4; float d = dinv[i];
    aggr[idx] = d * d * h[idx];                 // self-loop term
  }
}
__global__ __launch_bounds__(256) void k_aggr_edges(const int* rowI, const int* colI,
                                                    const float* dinv, const float* h,
                                                    float* aggr, int E) {
  int t = blockIdx.x * 256 + threadIdx.x;
  if (t >= E * 4) return;
  int e = t >> 2, c = (t & 3) * 4;
  int r = rowI[e], cl = colI[e];
  float norm = dinv[r] * dinv[cl];
  const float* hp = h + (size_t)cl * 16 + c;
  float* ap = aggr + (size_t)r * 16 + c;
#pragma unroll
  for (int j = 0; j < 4; ++j) atomicAdd(&ap[j], norm * hp[j]);
}

// ---------------- mlp2: aggr[N,16] -> xn[N,8] ----------------
__global__ __launch_bounds__(256) void k_mlp2(const float* __restrict__ aggr, MLP3 p,
                                              float* __restrict__ xn, int N) {
  __shared__ float lds[8][2][16 * 32];
  const int wave = threadIdx.x >> 5, lane = threadIdx.x & 31;
  const int rowBase = blockIdx.x * 128 + wave * 16;
  if (rowBase >= N) return;
  float* bufA = lds[wave][0]; float* bufB = lds[wave][1];
  const bool full = (rowBase + 16 <= N);
  if (full) {
    // 16 rows x 16 floats = 1024B contiguous: 2 x async b128
    const unsigned lb = ldsOffsetOf(bufA);
#pragma unroll
    for (int q = 0; q < 2; ++q) {
      int t = q * 32 + lane;                        // 16-byte chunk index
      int row = t >> 2, colb = (t & 3) * 16;
      async_load_b128(lb + (unsigned)(row * 128 + colb),
                      (const char*)(aggr + (size_t)rowBase * 16) + t * 16);
    }
  } else {
    for (int idx = lane; idx < 16 * 16; idx += 32) {
      int m = idx >> 4, c = idx & 15, r = rowBase + m;
      bufA[m * 32 + c] = (r < N) ? aggr[r * 16 + c] : 0.f;
    }
  }
  lds_sync_wave();
  async_sync_wave();
  tile_layer(bufA, 32, 16, 16, p.W1, p.b1, 32, bufB, 32, true,  lane);
  tile_layer(bufB, 32, 32, 32, p.W2, p.b2, 16, bufA, 32, true,  lane);
  tile_layer(bufA, 32, 16, 16, p.W3, p.b3, 8,  bufB, 32, false, lane);
  if (full) {
    // 16 rows x 8 floats = 512B contiguous: 1 x async b128
    const unsigned lb = ldsOffsetOf(bufB);
    int row = lane >> 1, colb = (lane & 1) * 16;
    async_store_b128((char*)(xn + (size_t)rowBase * 8) + lane * 16,
                     lb + (unsigned)(row * 128 + colb));
    async_sync_wave();
  } else {
    for (int idx = lane; idx < 16 * 8; idx += 32) {
      int m = idx >> 3, c = idx & 7, r = rowBase + m;
      if (r < N) xn[r * 8 + c] = bufB[m * 32 + c];
    }
  }
}

// ---------------- per-graph mean of xn ----------------
__global__ __launch_bounds__(64) void k_gmean(const float* xn, float* gm, int NPG) {
  __shared__ float s[64];
  int g = blockIdx.x, t = threadIdx.x, c = t & 7, sr = t >> 3;
  float sum = 0.f;
  for (int r = sr; r < NPG; r += 8) sum += xn[((size_t)g * NPG + r) * 8 + c];
  s[t] = sum; __syncthreads();
  for (int st = 32; st >= 8; st >>= 1) { if (t < st) s[t] += s[t + st]; __syncthreads(); }
  if (t < 8) gm[g * 8 + t] = s[t] / (float)NPG;
}

// ---------------- mlp_dag: gm[G,8] -> y[G,8] (partial tiles: scalar path) --------
__global__ __launch_bounds__(256) void k_dag(const float* __restrict__ gm, MLP3 p,
                                             float* __restrict__ y, int G) {
  __shared__ float lds[8][2][16 * 32];
  const int wave = threadIdx.x >> 5, lane = threadIdx.x & 31;
  const int rowBase = blockIdx.x * 128 + wave * 16;
  if (rowBase >= G) return;
  float* bufA = lds[wave][0]; float* bufB = lds[wave][1];
  for (int idx = lane; idx < 16 * 8; idx += 32) {
    int m = idx >> 3, c = idx & 7, r = rowBase + m;
    bufA[m * 32 + c] = (r < G) ? gm[r * 8 + c] : 0.f;
  }
  lds_sync_wave();
  tile_layer(bufA, 32, 8,  8,  p.W1, p.b1, 32, bufB, 32, true,  lane);
  tile_layer(bufB, 32, 32, 32, p.W2, p.b2, 16, bufA, 32, true,  lane);
  tile_layer(bufA, 32, 16, 16, p.W3, p.b3, 8,  bufB, 32, false, lane);
  for (int idx = lane; idx < 16 * 8; idx += 32) {
    int m = idx >> 3, c = idx & 7, r = rowBase + m;
    if (r < G) y[r * 8 + c] = bufB[m * 32 + c];
  }
}

// ---------------- global embedding z (tiny; scalar MLP) ----------------
__global__ __launch_bounds__(256) void k_zglobal(const float* y, int G, MLP3 p, float* z) {
  __shared__ float acc[256]; __shared__ float ym[8];
  __shared__ float h1[32];   __shared__ float h2[16];
  int t = threadIdx.x, c = t & 7, sr = t >> 3;
  float sum = 0.f;
  for (int r = sr; r < G; r += 32) sum += y[r * 8 + c];
  acc[t] = sum; __syncthreads();
  for (int st = 128; st >= 8; st >>= 1) { if (t < st) acc[t] += acc[t + st]; __syncthreads(); }
  if (t < 8) ym[t] = acc[t] / (float)G;
  __syncthreads();
  if (t < 32) { float s = p.b1[t]; for (int k = 0; k < 8;  ++k) s += ym[k] * p.W1[k*32+t]; h1[t] = fmaxf(s, 0.f); }
  __syncthreads();
  if (t < 16) { float s = p.b2[t]; for (int k = 0; k < 32; ++k) s += h1[k] * p.W2[k*16+t]; h2[t] = fmaxf(s, 0.f); }
  __syncthreads();
  if (t < 8)  { float s = p.b3[t]; for (int k = 0; k < 16; ++k) s += h2[k] * p.W3[k*8+t];  z[t] = s; }
}

// ---------------- mlp_op: [xn | y[batch] | z] (24) -> logit, masked ----------------
__global__ __launch_bounds__(256) void k_op(const float* __restrict__ xn,
                                            const float* __restrict__ y,
                                            const float* __restrict__ z,
                                            const int* __restrict__ op_msk, MLP3 p,
                                            float* __restrict__ logits, int N, int NPG) {
  __shared__ float lds[8][2][16 * 32];
  const int wave = threadIdx.x >> 5, lane = threadIdx.x & 31;
  const int rowBase = blockIdx.x * 128 + wave * 16;
  if (rowBase >= N) return;
  float* bufA = lds[wave][0]; float* bufB = lds[wave][1];
  const bool full = (rowBase + 16 <= N);
  if (full) {
    // xn part: 16 rows x 8 floats = 512B contiguous -> 1 async b128
    const unsigned lb = ldsOffsetOf(bufA);
    int row = lane >> 1, colb = (lane & 1) * 16;
    async_load_b128(lb + (unsigned)(row * 128 + colb),
                    (const char*)(xn + (size_t)rowBase * 8) + lane * 16);
    // y / z parts: cols 8..23
    for (int idx = lane; idx < 16 * 16; idx += 32) {
      int m = idx >> 4, c = (idx & 15) + 8, r = rowBase + m;
      float v = (c < 16) ? y[(r / NPG) * 8 + (c - 8)] : z[c - 16];
      bufA[m * 32 + c] = v;
    }
  } else {
    for (int idx = lane; idx < 16 * 24; idx += 32) {
      int m = idx / 24, c = idx % 24, r = rowBase + m;
      float v = 0.f;
      if (r < N) {
        if (c < 8)       v = xn[r * 8 + c];
        else if (c < 16) v = y[(r / NPG) * 8 + (c - 8)];
        else             v = z[c - 16];
      }
      bufA[m * 32 + c] = v;
    }
  }
  lds_sync_wave();
  async_sync_wave();
  tile_layer(bufA, 32, 24, 24, p.W1, p.b1, 32, bufB, 32, true, lane);
  tile_layer(bufB, 32, 32, 32, p.W2, p.b2, 16, bufA, 32, true, lane);
  if (lane < 16) {                                   // final 16 -> 1 layer
    int r = rowBase + lane;
    if (r < N) {
      float s = p.b3[0];
      for (int k = 0; k < 16; ++k) s += bufA[lane * 32 + k] * p.W3[k];
      s -= (1.0f - (float)op_msk[r]) * 1000.0f;
      logits[r] = s;
    }
  }
}

// ---------------- mlp_prlvl: [limit | y[g] | z] (17) -> logit, masked ----------------
__global__ __launch_bounds__(256) void k_prlvl(const float* __restrict__ y,
                                               const float* __restrict__ z,
                                               const int* __restrict__ msk, MLP3 p,
                                               float* __restrict__ logits, int R, int W) {
  __shared__ float lds[8][2][16 * 32];
  const int wave = threadIdx.x >> 5, lane = threadIdx.x & 31;
  const int rowBase = blockIdx.x * 128 + wave * 16;
  if (rowBase >= R) return;
  float* bufA = lds[wave][0]; float* bufB = lds[wave][1];
  for (int idx = lane; idx < 16 * 20; idx += 32) {   // 17 real cols, pad to 20
    int m = idx / 20, c = idx % 20, r = rowBase + m;
    float v = 0.f;
    if (r < R && c < 17) {
      int g = r / W, w = r % W;
      if (c == 0)     v = (float)(w + 1);
      else if (c < 9) v = y[g * 8 + (c - 1)];
      else            v = z[c - 9];
    }
    bufA[m * 32 + c] = v;
  }
  lds_sync_wave();
  tile_layer(bufA, 32, 20, 17, p.W1, p.b1, 32, bufB, 32, true, lane);
  tile_layer(bufB, 32, 32, 32, p.W2, p.b2, 16, bufA, 32, true, lane);
  if (lane < 16) {
    int r = rowBase + lane;
    if (r < R) {
      float s = p.b3[0];
      for (int k = 0; k < 16; ++k) s += bufA[lane * 32 + k] * p.W3[k];
      s -= (1.0f - (float)msk[r]) * 1000.0f;
      logits[r] = s;
    }
  }
}

// ---------------- global softmax over N (two-level) ----------------
__global__ __launch_bounds__(256) void k_rmax(const float* v, int n, float* part) {
  __shared__ float s[256];
  float m = -__builtin_inff();
  for (int i = blockIdx.x * 256 + threadIdx.x; i < n; i += 256 * gridDim.x) m = fmaxf(m, v[i]);
  s[threadIdx.x] = m; __syncthreads();
  for (int st = 128; st > 0; st >>= 1) { if (threadIdx.x < st) s[threadIdx.x] = fmaxf(s[threadIdx.x], s[threadIdx.x + st]); __syncthreads(); }
  if (threadIdx.x == 0) part[blockIdx.x] = s[0];
}
__global__ __launch_bounds__(256) void k_rfin(const float* part, float* out, int mode) {
  __shared__ float s[256];
  s[threadIdx.x] = part[threadIdx.x]; __syncthreads();
  for (int st = 128; st > 0; st >>= 1) {
    if (threadIdx.x < st) s[threadIdx.x] = mode ? (s[threadIdx.x] + s[threadIdx.x + st])
                                               : fmaxf(s[threadIdx.x], s[threadIdx.x + st]);
    __syncthreads();
  }
  if (threadIdx.x == 0) *out = s[0];
}
__global__ __launch_bounds__(256) void k_expsum(const float* v, int n, const float* mx,
                                                float* e_out, float* part) {
  __shared__ float s[256];
  float m = *mx, sum = 0.f;
  for (int i = blockIdx.x * 256 + threadIdx.x; i < n; i += 256 * gridDim.x) {
    float e = expf(v[i] - m); e_out[i] = e; sum += e;
  }
  s[threadIdx.x] = sum; __syncthreads();
  for (int st = 128; st > 0; st >>= 1) { if (threadIdx.x < st) s[threadIdx.x] += s[threadIdx.x + st]; __syncthreads(); }
  if (threadIdx.x == 0) part[blockIdx.x] = s[0];
}
__global__ __launch_bounds__(256) void k_norm(float* v, int n, const float* sum) {
  int i = blockIdx.x * 256 + threadIdx.x;
  if (i < n) v[i] = v[i] / (*sum);
}

// ---------------- per-graph softmax over W ----------------
__global__ __launch_bounds__(64) void k_prsoftmax(const float* logits, float* out, int W) {
  __shared__ float s[64];
  int g = blockIdx.x, t = threadIdx.x;
  float v = (t < W) ? logits[g * W + t] : -__builtin_inff();
  s[t] = v; __syncthreads();
  for (int st = 32; st > 0; st >>= 1) { if (t < st) s[t] = fmaxf(s[t], s[t + st]); __syncthreads(); }
  float mx = s[0]; __syncthreads();
  float e = (t < W) ? expf(v - mx) : 0.f;
  s[t] = e; __syncthreads();
  for (int st = 32; st > 0; st >>= 1) { if (t < st) s[t] += s[t + st]; __syncthreads(); }
  if (t < W) out[g * W + t] = e / s[0];
}

extern "C" void kernel_launch(void* const* d_in, const int* in_sizes, int n_in,
                              void* d_out, int out_size, void* d_ws, size_t ws_size,
                              hipStream_t stream) {
  const float* x      = (const float*)d_in[0];
  const int*   ei     = (const int*)d_in[1];
  const int*   op_msk = (const int*)d_in[3];
  const int*   pr_msk = (const int*)d_in[4];

  const int N  = in_sizes[3];
  const int E  = in_sizes[1] / 2;
  const int GW = in_sizes[4];
  const int W  = 50;            // reference num_workers
  const int G  = GW / W;
  const int NPG = N / G;        // sorted equal segments per reference
  const int* rowI = ei;
  const int* colI = ei + E;

  auto mk = [&](int base) {
    MLP3 m; m.W1 = (const float*)d_in[base + 0]; m.b1 = (const float*)d_in[base + 1];
            m.W2 = (const float*)d_in[base + 2]; m.b2 = (const float*)d_in[base + 3];
            m.W3 = (const float*)d_in[base + 4]; m.b3 = (const float*)d_in[base + 5];
    return m;
  };
  MLP3 m1 = mk(6), m2 = mk(12), mdag = mk(18), mglob = mk(24), mop = mk(30), mpr = mk(36);

  float* base = (float*)d_ws;
  size_t o = 0;
  float* h     = base + o; o += (size_t)N * 16;
  float* deg   = base + o; o += N;
  float* dinv  = base + o; o += N;
  float* aggr  = base + o; o += (size_t)N * 16;
  float* xn    = base + o; o += (size_t)N * 8;
  float* gm    = base + o; o += (size_t)G * 8;
  float* yv    = base + o; o += (size_t)G * 8;
  float* zv    = base + o; o += 8;
  float* oplog = base + o; o += N;
  float* prlog = base + o; o += GW;
  float* pmax  = base + o; o += 256;
  float* psum  = base + o; o += 256;
  float* scal  = base + o; o += 4;

  float* outOps = (float*)d_out;
  float* outPr  = outOps + N;

  k_deg_init <<<(N + 255) / 256, 256, 0, stream>>>(deg, N);
  k_deg_edges<<<(E + 255) / 256, 256, 0, stream>>>(colI, deg, E);
  k_dinv     <<<(N + 255) / 256, 256, 0, stream>>>(deg, dinv, N);
  k_mlp1     <<<(N + 127) / 128, 256, 0, stream>>>(x, m1, h, N);
  k_aggr_init<<<(N * 16 + 255) / 256, 256, 0, stream>>>(h, dinv, aggr, N);
  k_aggr_edges<<<(E * 4 + 255) / 256, 256, 0, stream>>>(rowI, colI, dinv, h, aggr, E);
  k_mlp2     <<<(N + 127) / 128, 256, 0, stream>>>(aggr, m2, xn, N);
  k_gmean    <<<G, 64, 0, stream>>>(xn, gm, NPG);
  k_dag      <<<(G + 127) / 128, 256, 0, stream>>>(gm, mdag, yv, G);
  k_zglobal  <<<1, 256, 0, stream>>>(yv, G, mglob, zv);
  k_op       <<<(N + 127) / 128, 256, 0, stream>>>(xn, yv, zv, op_msk, mop, oplog, N, NPG);
  k_prlvl    <<<(GW + 127) / 128, 256, 0, stream>>>(yv, zv, pr_msk, mpr, prlog, GW, W);
  k_rmax     <<<256, 256, 0, stream>>>(oplog, N, pmax);
  k_rfin     <<<1, 256, 0, stream>>>(pmax, &scal[0], 0);
  k_expsum   <<<256, 256, 0, stream>>>(oplog, N, &scal[0], outOps, psum);
  k_rfin     <<<1, 256, 0, stream>>>(psum, &scal[1], 1);
  k_norm     <<<(N + 255) / 256, 256, 0, stream>>>(outOps, N, &scal[1]);
  k_prsoftmax<<<G, 64, 0, stream>>>(prlog, outPr, W);
}